// GraphNorm_9887014715653
// MI455X (gfx1250) — compile-verified
//
#include <hip/hip_runtime.h>
#include <hip/hip_bf16.h>

#define NUM_SEGMENTS 1024
#define CH 128
#define EPS 1e-5f
#define NPB 1024   // nodes per block in the stats pass

typedef __attribute__((ext_vector_type(2))) float v2f;
typedef __attribute__((ext_vector_type(4))) float v4f;
typedef __attribute__((ext_vector_type(8))) float v8f;

// Workspace layout (32-bit words):
//  [0,                  1024*128)  sum    (becomes mean after finalize)
//  [1024*128,         2*1024*128)  sumsq  (becomes inv_std after finalize)
//  [2*1024*128,   2*1024*128+1024) seg_start (i32)
//  [.. +1024,          .. +2048)   seg_end   (i32)
#define SUM_OFF   0
#define SQ_OFF    (NUM_SEGMENTS * CH)
#define START_OFF (2 * NUM_SEGMENTS * CH)
#define END_OFF   (2 * NUM_SEGMENTS * CH + NUM_SEGMENTS)
#define WS_WORDS  (2 * NUM_SEGMENTS * CH + 2 * NUM_SEGMENTS)

__global__ void gn_zero_ws(unsigned* ws) {
    int i = blockIdx.x * blockDim.x + threadIdx.x;
    if (i < WS_WORDS) ws[i] = 0u;
}

// batch is sorted: segment b occupies [start[b], end[b]); unwritten (empty) segs stay 0/0.
__global__ void gn_bounds(const int* __restrict__ batch, int* __restrict__ seg_start,
                          int* __restrict__ seg_end, int N) {
    int i = blockIdx.x * blockDim.x + threadIdx.x;
    if (i >= N) return;
    int b = batch[i];
    if (i == 0 || batch[i - 1] != b) seg_start[b] = i;
    if (i == N - 1 || batch[i + 1] != b) seg_end[b] = i + 1;
}

__device__ __forceinline__ void gn_flush(float* __restrict__ sum, float* __restrict__ sumsq,
                                         int seg, int ch, float aS, float aQ, int lane) {
    // lanes 0-15 and 16-31 hold duplicate channel sums; flush from the low half only.
    if (seg >= 0 && lane < 16) {
        __hip_atomic_fetch_add(&sum[seg * CH + ch], aS, __ATOMIC_RELAXED, __HIP_MEMORY_SCOPE_AGENT);
        __hip_atomic_fetch_add(&sumsq[seg * CH + ch], aQ, __ATOMIC_RELAXED, __HIP_MEMORY_SCOPE_AGENT);
    }
}

// Pass 1: per-segment sum and sum-of-squares.
// Block = 256 threads = 8 waves; wave w owns channels [16w, 16w+16).
// Each wave walks 4 nodes per iteration; uniform-segment groups are reduced with
// V_WMMA_F32_16X16X4_F32 (A = ones(16x4), B = 4x16 fp32 tile -> column sums in C,
// channel n = lane%16 in every C VGPR). fp32 MACs, so exact fp32-add semantics.
__global__ void gn_stats(const float* __restrict__ x, const int* __restrict__ batch,
                         float* __restrict__ sum, float* __restrict__ sumsq, int N) {
    const int lane = threadIdx.x & 31;
    const int wave = threadIdx.x >> 5;                 // 0..7
    const int ch = (wave << 4) + (lane & 15);          // this lane's channel
    const int rowoff = (lane >> 4) << 1;               // lanes 0-15 -> rows n,n+1 ; 16-31 -> n+2,n+3
    const int base = blockIdx.x * NPB;

    float accS = 0.0f, accQ = 0.0f;
    int curseg = -1;
    v2f ones; ones.x = 1.0f; ones.y = 1.0f;
    v8f zero8 = {0.f, 0.f, 0.f, 0.f, 0.f, 0.f, 0.f, 0.f};

    for (int g = 0; g < NPB; g += 4) {
        const int n = base + g;
        if (n >= N) break;
        const bool full = (n + 3 < N);
        const int b0 = batch[n];
        const int b3 = full ? batch[n + 3] : -2;       // sorted => b0==b3 implies uniform group

        if (full && b0 == b3) {
            if (b0 != curseg) {
                gn_flush(sum, sumsq, curseg, ch, accS, accQ, lane);
                accS = 0.0f; accQ = 0.0f; curseg = b0;
            }
            const float* p = x + (size_t)(n + rowoff) * CH + ch;
            float v0 = p[0];
            float v1 = p[CH];
            v2f bS; bS.x = v0;      bS.y = v1;
            v2f bQ; bQ.x = v0 * v0; bQ.y = v1 * v1;
            v8f cs = __builtin_amdgcn_wmma_f32_16x16x4_f32(
                false, ones, false, bS, (short)0, zero8, false, false);
            v8f cq = __builtin_amdgcn_wmma_f32_16x16x4_f32(
                false, ones, false, bQ, (short)0, zero8, false, false);
            accS += cs[0];
            accQ += cq[0];
        } else {
            // segment boundary inside the 4-group, or tail: per-node scalar path
            for (int j = 0; j < 4; ++j) {
                int m = n + j;
                if (m >= N) break;
                int b = batch[m];
                if (b != curseg) {
                    gn_flush(sum, sumsq, curseg, ch, accS, accQ, lane);
                    accS = 0.0f; accQ = 0.0f; curseg = b;
                }
                float v = x[(size_t)m * CH + ch];
                accS += v;
                accQ += v * v;
            }
        }
    }
    gn_flush(sum, sumsq, curseg, ch, accS, accQ, lane);
}

// In-place: sum -> mean, sumsq -> inv_std.
__global__ void gn_finalize(float* __restrict__ sum, float* __restrict__ sumsq,
                            const int* __restrict__ seg_start, const int* __restrict__ seg_end) {
    int b = blockIdx.x;
    int c = threadIdx.x;
    float cnt = (float)(seg_end[b] - seg_start[b]);
    cnt = fmaxf(cnt, 1.0f);                            // matches reference's count clamp
    float icnt = 1.0f / cnt;
    float m = sum[b * CH + c] * icnt;
    float q = sumsq[b * CH + c] * icnt;
    float var = fmaxf(q - m * m, 0.0f);
    sum[b * CH + c] = m;
    sumsq[b * CH + c] = 1.0f / sqrtf(var + EPS);
}

// Pass 2: out = (x - mean[b]) * inv_std[b] * weight + bias, B128 vectorized.
// One wave covers one 512B node row; stats reads hit L2 (1 MB resident).
__global__ void gn_normalize(const float* __restrict__ x, const int* __restrict__ batch,
                             const float* __restrict__ mean, const float* __restrict__ invstd,
                             const float* __restrict__ weight, const float* __restrict__ bias,
                             float* __restrict__ out, int N) {
    int idx = blockIdx.x * blockDim.x + threadIdx.x;   // one thread = 4 channels
    int total = N * (CH / 4);
    if (idx >= total) return;
    int n = idx >> 5;                                  // 32 threads per node row
    int c4 = (idx & 31) << 2;
    int b = batch[n];
    const v4f xv = *(const v4f*)(x + (size_t)n * CH + c4);
    const v4f mv = *(const v4f*)(mean + b * CH + c4);
    const v4f iv = *(const v4f*)(invstd + b * CH + c4);
    const v4f wv = *(const v4f*)(weight + c4);
    const v4f bv = *(const v4f*)(bias + c4);
    v4f o = (xv - mv) * iv * wv + bv;
    *(v4f*)(out + (size_t)n * CH + c4) = o;
}

extern "C" void kernel_launch(void* const* d_in, const int* in_sizes, int n_in,
                              void* d_out, int out_size, void* d_ws, size_t ws_size,
                              hipStream_t stream) {
    const float* x      = (const float*)d_in[0];
    const int*   batch  = (const int*)d_in[1];
    const float* weight = (const float*)d_in[2];
    const float* bias   = (const float*)d_in[3];
    float* out = (float*)d_out;

    const int N = in_sizes[0] / CH;

    unsigned* wsw   = (unsigned*)d_ws;
    float* sum      = (float*)d_ws + SUM_OFF;
    float* sumsq    = (float*)d_ws + SQ_OFF;
    int* seg_start  = (int*)d_ws + START_OFF;
    int* seg_end    = (int*)d_ws + END_OFF;

    gn_zero_ws<<<(WS_WORDS + 255) / 256, 256, 0, stream>>>(wsw);
    gn_bounds<<<(N + 255) / 256, 256, 0, stream>>>(batch, seg_start, seg_end, N);
    gn_stats<<<(N + NPB - 1) / NPB, 256, 0, stream>>>(x, batch, sum, sumsq, N);
    gn_finalize<<<NUM_SEGMENTS, CH, 0, stream>>>(sum, sumsq, seg_start, seg_end);
    {
        int total = N * (CH / 4);
        gn_normalize<<<(total + 255) / 256, 256, 0, stream>>>(
            x, batch, sum, sumsq, weight, bias, out, N);
    }
}